// SparseConv3d_85392539779123
// MI455X (gfx1250) — compile-verified
//
#include <hip/hip_runtime.h>

// Problem constants (from reference): B=2, CI=32, CO=64, K=3, D=H=W=96
#define BB 2
#define CI 32
#define CO 64
#define KS 3
#define DD 96
#define HH 96
#define WW 96

#define WTILE 32                    // W positions per block (2 sub-tiles of 16 per wave)
#define NSUB (WTILE / 16)           // 2 accumulators per wave
#define PATCH_W (WTILE + KS - 1)    // 34
#define PATCH_WP 36                 // padded LDS stride
#define KRED (CI * KS * KS)         // 288 (ci,kd,kh) steps; kw in the K=4 quad
#define NTHREADS 128                // 4 waves: one 16-CO tile per wave

typedef float v2f __attribute__((ext_vector_type(2)));
typedef float v8f __attribute__((ext_vector_type(8)));

__global__ __launch_bounds__(NTHREADS)
void sparse_conv3d_wmma_f32(const float* __restrict__ x,
                            const int*   __restrict__ mask,
                            const float* __restrict__ w,
                            const float* __restrict__ bias,
                            float*       __restrict__ out)
{
    __shared__ float patch[CI * KS * KS * PATCH_WP];   // 41472 B

    const int tid  = threadIdx.x;
    const int lane = tid & 31;
    const int wave = tid >> 5;

    // blockIdx.x = (((b*D + d)*H + h)*(W/WTILE) + wt)
    int idx = blockIdx.x;
    const int wt = idx % (WW / WTILE); idx /= (WW / WTILE);
    const int h  = idx % HH;           idx /= HH;
    const int d  = idx % DD;           idx /= DD;
    const int b  = idx;
    const int w0 = wt * WTILE;

    // ---- Stage input halo patch [CI][3][3][34] into LDS (zero-padded) ----
    for (int i = tid; i < CI * KS * KS * PATCH_W; i += NTHREADS) {
        int wx = i % PATCH_W;
        int t  = i / PATCH_W;
        int hy = t % KS; t /= KS;
        int dz = t % KS;
        int ci = t / KS;
        int gd = d + dz - 1, gh = h + hy - 1, gw = w0 + wx - 1;
        float v = 0.0f;
        if ((unsigned)gd < (unsigned)DD && (unsigned)gh < (unsigned)HH &&
            (unsigned)gw < (unsigned)WW) {
            v = x[(((size_t)(b * CI + ci) * DD + gd) * HH + gh) * WW + gw];
        }
        patch[((ci * KS + dz) * KS + hy) * PATCH_WP + wx] = v;
    }
    __syncthreads();

    // ---- Implicit GEMM per wave: M=16 COs, N=32 positions, K=864 ----
    const int  co_base = wave * 16;
    const int  m  = lane & 15;       // A: row M; B: column N (within sub-tile)
    const bool lo = lane < 16;       // lanes 0-15 hold K={0,1}; 16-31 hold K={2,3}
    const int  ka = lo ? 0 : 2;      // first K element this half-wave owns

    v8f acc0 = {}, acc1 = {};
    const float* wrow = w + (size_t)(co_base + m) * (CI * KS * KS * KS);

    for (int r = 0; r < KRED; ++r) {
        // A fragment: 16x4 f32 weights, K-quad = {kw0,kw1,kw2,0}.
        // All loads unconditional + in-bounds; zeroing via cndmask (no exec branches).
        const float* wr = wrow + r * KS;
        float wv0 = wr[ka];          // kw0 (lo) / kw2 (hi)
        float wv1 = wr[1];           // kw1 (always valid)
        v2f a;
        a.x = wv0;
        a.y = lo ? wv1 : 0.0f;       // hi half K=3 is the zero pad

        // B fragments from LDS: B[kw][n] = patch[r][n + kw + 16*sub]
        const float* pr = patch + r * PATCH_WP;
        float p0a = pr[m + ka];
        float p0b = pr[m + 1];
        float p1a = pr[16 + m + ka];
        float p1b = pr[16 + m + 1];
        v2f b0; b0.x = p0a; b0.y = lo ? p0b : 0.0f;
        v2f b1; b1.x = p1a; b1.y = lo ? p1b : 0.0f;

        // Two independent WMMA chains share one A fragment
        acc0 = __builtin_amdgcn_wmma_f32_16x16x4_f32(
            false, a, false, b0, (short)0, acc0, false, false);
        acc1 = __builtin_amdgcn_wmma_f32_16x16x4_f32(
            false, a, false, b1, (short)0, acc1, false, false);
    }

    // ---- Epilogue: bias add, mask multiply, store two 16x16 tiles ----
    // C/D layout: VGPR i -> M = i + (lane<16 ? 0 : 8); N = lane&15.
    const size_t plane = (size_t)DD * HH * WW;
    const int    mrow0 = lo ? 0 : 8;
    const size_t srow  = (size_t)d * HH * WW + (size_t)h * WW;

#pragma unroll
    for (int s = 0; s < NSUB; ++s) {
        v8f&      acc   = (s == 0) ? acc0 : acc1;
        const int gwout = w0 + s * 16 + m;
        const float fm  =
            (float)mask[((size_t)b * DD + d) * HH * WW + (size_t)h * WW + gwout];
        size_t obase = ((size_t)(b * CO + co_base + mrow0) * plane) + srow + gwout;
#pragma unroll
        for (int i = 0; i < 8; ++i) {
            float val = (acc[i] + bias[co_base + mrow0 + i]) * fm;
            out[obase + (size_t)i * plane] = val;
        }
    }
}

extern "C" void kernel_launch(void* const* d_in, const int* in_sizes, int n_in,
                              void* d_out, int out_size, void* d_ws, size_t ws_size,
                              hipStream_t stream)
{
    const float* x    = (const float*)d_in[0];
    const int*   mask = (const int*)  d_in[1];
    const float* w    = (const float*)d_in[2];
    const float* bias = (const float*)d_in[3];
    float*       out  = (float*)d_out;

    const int nblocks = BB * DD * HH * (WW / WTILE);  // 55296
    sparse_conv3d_wmma_f32<<<dim3(nblocks), dim3(NTHREADS), 0, stream>>>(
        x, mask, w, bias, out);
}